// GCN3_bias_icml2019_20727512170667
// MI455X (gfx1250) — compile-verified
//
#include <hip/hip_runtime.h>

typedef __attribute__((ext_vector_type(2))) float v2f;
typedef __attribute__((ext_vector_type(4))) float v4f;
typedef __attribute__((ext_vector_type(8))) float v8f;

#define DDIM 128
#define LAMBDA 0.001f

// ---------------------------------------------------------------------------
// Async global->LDS staging (CDNA5 GLOBAL_LOAD_ASYNC_TO_LDS_B128, ASYNCcnt),
// with a plain load/store fallback if the builtins aren't declared.
// Probe result: param 1 is `__attribute__((vector_size(16))) int __device__ *`
// (global/addrspace(1) pointer to 16-byte int vector, non-const).
// ---------------------------------------------------------------------------
#if __has_builtin(__builtin_amdgcn_global_load_async_to_lds_b128) && \
    __has_builtin(__builtin_amdgcn_s_wait_asynccnt)
#define HAVE_ASYNC_LDS 1
#else
#define HAVE_ASYNC_LDS 0
#endif

typedef int v4i_vs __attribute__((vector_size(16)));

__device__ __forceinline__ void stage16(const float* __restrict__ gsrc, float* ldst)
{
#if HAVE_ASYNC_LDS
    typedef __attribute__((address_space(1))) v4i_vs gv4i;
    typedef __attribute__((address_space(3))) v4i_vs lv4i;
    __builtin_amdgcn_global_load_async_to_lds_b128(
        (gv4i*)(uintptr_t)gsrc,
        (lv4i*)(unsigned int)(uintptr_t)ldst,
        /*offset=*/0, /*cpol=*/0);
#else
    *(v4f*)ldst = *(const v4f*)gsrc;
#endif
}

__device__ __forceinline__ void stage_join()
{
#if HAVE_ASYNC_LDS
    __builtin_amdgcn_s_wait_asynccnt(0);
#endif
    __syncthreads();
}

// ---------------------------------------------------------------------------
// SpMM: out[dst[e], :] += val[e] * X[src[e], :]   (one wave per edge, 4 f32/lane)
// ---------------------------------------------------------------------------
__global__ __launch_bounds__(256)
void spmm_edge_kernel(const int* __restrict__ srcIdx, const int* __restrict__ dstIdx,
                      const float* __restrict__ val, const float* __restrict__ X,
                      float* __restrict__ Out, int E)
{
    const long long t = (long long)blockIdx.x * 256 + threadIdx.x;
    const int e = (int)(t >> 5);
    if (e >= E) return;
    const int lane = (int)(t & 31);
    const int s = srcIdx[e];
    const int d = dstIdx[e];
    const float v = val[e];
    const v4f x = ((const v4f*)(X + (size_t)s * DDIM))[lane];
    float* o = Out + (size_t)d * DDIM + lane * 4;
    atomicAdd(o + 0, v * x[0]);
    atomicAdd(o + 1, v * x[1]);
    atomicAdd(o + 2, v * x[2]);
    atomicAdd(o + 3, v * x[3]);
}

// ---------------------------------------------------------------------------
// buf = relu(buf + emb * deg[row])
// ---------------------------------------------------------------------------
__global__ __launch_bounds__(256)
void relu_combine_kernel(float* __restrict__ buf, const float* __restrict__ emb,
                         const float* __restrict__ deg, long long n)
{
    const long long i = (long long)blockIdx.x * 256 + threadIdx.x;
    if (i >= n) return;
    const long long row = i >> 7;  // D = 128
    float v = buf[i] + emb[i] * deg[row];
    buf[i] = v > 0.f ? v : 0.f;
}

// ---------------------------------------------------------------------------
// out = w0*e0 + w1*g1 + w2*g2 ; accumulate sum(out^2) into ssq_acc
// ---------------------------------------------------------------------------
__global__ __launch_bounds__(256)
void combine3_ssq_kernel(const float* __restrict__ e0, const float* __restrict__ g1,
                         const float* __restrict__ g2, const float* __restrict__ add_w,
                         float* __restrict__ out, float* __restrict__ ssq_acc, long long n)
{
    __shared__ float red;
    if (threadIdx.x == 0) red = 0.f;
    __syncthreads();
    const long long i = (long long)blockIdx.x * 256 + threadIdx.x;
    float sq = 0.f;
    if (i < n) {
        const float w0 = add_w[0], w1 = add_w[1], w2 = add_w[2];
        const float v = w0 * e0[i] + w1 * g1[i] + w2 * g2[i];
        out[i] = v;
        sq = v * v;
    }
    #pragma unroll
    for (int m = 16; m; m >>= 1) sq += __shfl_xor(sq, m, 32);
    if ((threadIdx.x & 31) == 0) atomicAdd(&red, sq);
    __syncthreads();
    if (threadIdx.x == 0) atomicAdd(ssq_acc, red);
}

// ---------------------------------------------------------------------------
// WMMA GEMM: Out[M,N] = leaky( A[gather(M),K] @ W[K,N] + bias[N], slope )
//
// Block = 256 threads = 8 waves = one 128x16 macro-tile (8 M-tiles x 1 N-strip).
// The K x 16 weight strip is staged once into LDS (async-to-LDS when available)
// and shared by all 8 waves; each wave runs K/4 V_WMMA_F32_16X16X4_F32 ops with
// B fragments from LDS and A fragments via constant-stride global float2 loads.
// EXEC is all-1s everywhere: exact grid, uniform control flow.
// ---------------------------------------------------------------------------
__global__ __launch_bounds__(256)
void wmma_gemm_leaky_kernel(const float* __restrict__ A, const int* __restrict__ gatherIdx,
                            const float* __restrict__ W, const float* __restrict__ bias,
                            float* __restrict__ Out, int K, int N, float slope)
{
    __shared__ float Wlds[256 * 16];  // up to K=256, 16 columns (16 KB)

    const int tilesN = N >> 4;
    const int bm = blockIdx.x / tilesN;
    const int tn = blockIdx.x - bm * tilesN;
    const int n0 = tn * 16;

    // Cooperative staging of W[:, n0:n0+16] into LDS.
    // K*16 is a multiple of 1024 (K=128 or 256): all threads iterate equally.
    for (int base = threadIdx.x * 4; base < K * 16; base += 1024) {
        const int row = base >> 4;       // 16 floats per LDS row
        const int col = base & 15;       // 0,4,8,12
        stage16(W + (size_t)row * N + n0 + col, &Wlds[base]);
    }
    stage_join();

    const int lane = threadIdx.x & 31;
    const int l16  = lane & 15;
    const int half = lane >> 4;
    const int tm   = bm * 8 + (threadIdx.x >> 5);

    int row = tm * 16 + l16;             // A-matrix M index (both lane halves carry M=0..15)
    if (gatherIdx) row = gatherIdx[row];
    const float* ap = A + (size_t)row * K + half * 2;   // VGPR0/1 hold K = {ka, ka+1}
    int lo = half * 32 + l16;            // LDS index of B row ka, column l16

    v8f acc = {0.f, 0.f, 0.f, 0.f, 0.f, 0.f, 0.f, 0.f};

    for (int kb = 0; kb < K; kb += 4) {
        v2f a = *(const v2f*)ap;         // A 16x4 fragment (2 f32 / lane)
        ap += 4;
        v2f b;
        b.x = Wlds[lo];                  // B 4x16 fragment: rows ka, ka+1
        b.y = Wlds[lo + 16];
        lo += 64;
        acc = __builtin_amdgcn_wmma_f32_16x16x4_f32(
            /*neg_a=*/false, a, /*neg_b=*/false, b,
            /*c_mod=*/(short)0, acc, /*reuse_a=*/false, /*reuse_b=*/false);
    }

    const int n = n0 + l16;
    const float bs = bias[n];
    float* orow = Out + (size_t)(tm * 16 + half * 8) * N + n;   // C/D: VGPR r -> M = r + half*8
    #pragma unroll
    for (int r = 0; r < 8; ++r) {
        float v = acc[r] + bs;
        v = v > 0.f ? v : slope * v;
        orow[(size_t)r * N] = v;
    }
}

// ---------------------------------------------------------------------------
// pred_b = dot(H2[b], gi[item0[b]]) + ub[user0[b]] + ib[item0[b]] + avg
// accumulate (pred - rating)^2 into sse_acc. One wave per batch element.
// ---------------------------------------------------------------------------
__global__ __launch_bounds__(256)
void dot_loss_kernel(const float* __restrict__ H2, const float* __restrict__ gi,
                     const int* __restrict__ user0, const int* __restrict__ item0,
                     const float* __restrict__ ub, const float* __restrict__ ib,
                     const float* __restrict__ ratings, const float* __restrict__ avg,
                     float* __restrict__ sse_acc, int B)
{
    __shared__ float red;
    if (threadIdx.x == 0) red = 0.f;
    __syncthreads();
    const long long t = (long long)blockIdx.x * 256 + threadIdx.x;
    const int b = (int)(t >> 5);
    const int lane = (int)(t & 31);
    float contrib = 0.f;
    if (b < B) {
        const int it = item0[b];
        const v4f hv = ((const v4f*)(H2 + (size_t)b * DDIM))[lane];
        const v4f gv = ((const v4f*)(gi + (size_t)it * DDIM))[lane];
        float s = hv[0] * gv[0] + hv[1] * gv[1] + hv[2] * gv[2] + hv[3] * gv[3];
        #pragma unroll
        for (int m = 16; m; m >>= 1) s += __shfl_xor(s, m, 32);
        if (lane == 0) {
            const float pred = s + ub[user0[b]] + ib[it] + avg[0];
            const float d = pred - ratings[b];
            contrib = d * d;
        }
    }
    if ((threadIdx.x & 31) == 0) atomicAdd(&red, contrib);
    __syncthreads();
    if (threadIdx.x == 0) atomicAdd(sse_acc, red);
}

__global__ void finalize_kernel(const float* __restrict__ acc, float* __restrict__ out,
                                float invB, float invUD, float invID)
{
    const float loss2 = acc[0] * invB;
    const float l2 = LAMBDA * (acc[1] * invUD + acc[2] * invID);
    out[0] = loss2 + l2;
    out[1] = loss2;
}

// ---------------------------------------------------------------------------
extern "C" void kernel_launch(void* const* d_in, const int* in_sizes, int n_in,
                              void* d_out, int out_size, void* d_ws, size_t ws_size,
                              hipStream_t stream)
{
    const int*   edge_u   = (const int*)  d_in[0];
    const int*   edge_i   = (const int*)  d_in[1];
    const float* edge_val = (const float*)d_in[2];
    const float* d_i      = (const float*)d_in[3];
    const float* d_j      = (const float*)d_in[4];
    const int*   user0    = (const int*)  d_in[5];
    const int*   item_i0  = (const int*)  d_in[6];
    const float* ratings  = (const float*)d_in[7];
    const float* avg_rat  = (const float*)d_in[8];
    const float* eu       = (const float*)d_in[9];
    const float* ei       = (const float*)d_in[10];
    const float* add_w    = (const float*)d_in[11];
    const float* w1       = (const float*)d_in[12];
    const float* b1       = (const float*)d_in[13];
    const float* w2       = (const float*)d_in[14];
    const float* b2       = (const float*)d_in[15];
    const float* ub       = (const float*)d_in[16];
    const float* ib       = (const float*)d_in[17];

    const int E = in_sizes[0];
    const int U = in_sizes[3];
    const int I = in_sizes[4];
    const int B = in_sizes[5];
    const long long UD = (long long)U * DDIM;
    const long long ID = (long long)I * DDIM;

    // workspace layout (floats)
    float* ws    = (float*)d_ws;
    float* bufAU = ws;                               // agg_u -> g1u -> gu   [U*D]
    float* bufBU = bufAU + UD;                       // agg_u2 -> g2u        [U*D]
    float* bufAI = bufBU + UD;                       // agg_i -> g1i -> gi   [I*D]
    float* bufBI = bufAI + ID;                       // agg_i2 -> g2i        [I*D]
    float* H1    = bufBI + ID;                       // [B, 2D]
    float* H2    = H1 + (long long)B * 2 * DDIM;     // [B, D]
    float* acc   = H2 + (long long)B * DDIM;         // [0]=sse [1]=ssq_gu [2]=ssq_gi

    const int spmmBlocks = (int)(((long long)E * 32 + 255) / 256);
    const int uBlocks    = (int)((UD + 255) / 256);
    const int iBlocks    = (int)((ID + 255) / 256);

    // zero accumulated buffers (every call: deterministic, graph-capture safe)
    (void)hipMemsetAsync(bufAU, 0, (size_t)UD * 4, stream);
    (void)hipMemsetAsync(bufAI, 0, (size_t)ID * 4, stream);
    (void)hipMemsetAsync(acc,   0, 4 * sizeof(float), stream);

    // Layer 1: aggregate then relu-combine with degree-scaled self connection
    spmm_edge_kernel<<<spmmBlocks, 256, 0, stream>>>(edge_i, edge_u, edge_val, ei, bufAU, E); // spmm_ui(ei)
    spmm_edge_kernel<<<spmmBlocks, 256, 0, stream>>>(edge_u, edge_i, edge_val, eu, bufAI, E); // spmm_iu(eu)
    relu_combine_kernel<<<uBlocks, 256, 0, stream>>>(bufAU, eu, d_i, UD);  // g1u
    relu_combine_kernel<<<iBlocks, 256, 0, stream>>>(bufAI, ei, d_j, ID);  // g1i

    // Layer 2
    (void)hipMemsetAsync(bufBU, 0, (size_t)UD * 4, stream);
    (void)hipMemsetAsync(bufBI, 0, (size_t)ID * 4, stream);
    spmm_edge_kernel<<<spmmBlocks, 256, 0, stream>>>(edge_i, edge_u, edge_val, bufAI, bufBU, E); // spmm_ui(g1i)
    spmm_edge_kernel<<<spmmBlocks, 256, 0, stream>>>(edge_u, edge_i, edge_val, bufAU, bufBI, E); // spmm_iu(g1u)
    relu_combine_kernel<<<uBlocks, 256, 0, stream>>>(bufBU, bufAU, d_i, UD); // g2u
    relu_combine_kernel<<<iBlocks, 256, 0, stream>>>(bufBI, bufAI, d_j, ID); // g2i

    // Layer mix + fused L2-norm reduction (gu/gi written in place over g1u/g1i)
    combine3_ssq_kernel<<<uBlocks, 256, 0, stream>>>(eu, bufAU, bufBU, add_w, bufAU, acc + 1, UD);
    combine3_ssq_kernel<<<iBlocks, 256, 0, stream>>>(ei, bufAI, bufBI, add_w, bufAI, acc + 2, ID);

    // MLP via f32 WMMA: [B,128]@[128,256] -> leaky -> [B,256]@[256,128] -> leaky
    {
        const int blocks1 = (B / 128) * (2 * DDIM / 16);   // 128x16 macro-tiles
        wmma_gemm_leaky_kernel<<<blocks1, 256, 0, stream>>>(bufAU, user0, w1, b1, H1,
                                                            DDIM, 2 * DDIM, 0.1f);
        const int blocks2 = (B / 128) * (DDIM / 16);
        wmma_gemm_leaky_kernel<<<blocks2, 256, 0, stream>>>(H1, nullptr, w2, b2, H2,
                                                            2 * DDIM, DDIM, 0.1f);
    }

    // Prediction + MSE accumulation
    const int dotBlocks = (int)(((long long)B * 32 + 255) / 256);
    dot_loss_kernel<<<dotBlocks, 256, 0, stream>>>(H2, bufAI, user0, item_i0, ub, ib,
                                                   ratings, avg_rat, acc, B);

    finalize_kernel<<<1, 1, 0, stream>>>(acc, (float*)d_out,
                                         1.f / (float)B,
                                         1.f / (float)UD,
                                         1.f / (float)ID);
}